// DeformableTransformerDecoder_56667798504008
// MI455X (gfx1250) — compile-verified
//
#include <hip/hip_runtime.h>
#include <hip/hip_bf16.h>

// ---------------------------------------------------------------------------
// Problem constants (from reference): BS,NV,Q,C,H,W = 2,2,131072,128,128,240
// SCALE lo=(-8,-8,2), hi-lo=(16,16,32)
// ---------------------------------------------------------------------------
namespace {
constexpr int kBS = 2;
constexpr int kNV = 2;
constexpr int kQ  = 131072;   // 2^17
constexpr int kC  = 128;
constexpr int kH  = 128;
constexpr int kW  = 240;
constexpr int kP  = kH * kW;  // 30720 pixels per image
constexpr size_t kOffCenter = (size_t)kBS * kQ * kC;                    // features size
constexpr size_t kOffValid  = kOffCenter + (size_t)kBS * kNV * kQ * 2; // + center_im
}

// Native vector types (clang ext vectors — required by the amdgcn builtins)
typedef int   v4i __attribute__((ext_vector_type(4)));
typedef float v4f __attribute__((ext_vector_type(4)));
typedef __attribute__((address_space(1))) v4i g_v4i;
typedef __attribute__((address_space(3))) v4i l_v4i;

#if defined(__has_builtin)
#if __has_builtin(__builtin_amdgcn_global_load_async_to_lds_b128) && \
    __has_builtin(__builtin_amdgcn_s_wait_asynccnt)
#define DTD_USE_ASYNC_BUILTIN 1
#endif
#endif

// ---------------------------------------------------------------------------
// CDNA5 async global->LDS copy (ASYNCcnt-tracked). Builtin if present, else
// inline asm using the gfx1250 mnemonic (CDNA5 ISA).
// ---------------------------------------------------------------------------
__device__ __forceinline__ void dtd_async_copy_b128(const float* g, float* l) {
#ifdef DTD_USE_ASYNC_BUILTIN
  __builtin_amdgcn_global_load_async_to_lds_b128(
      (g_v4i*)g, (l_v4i*)l, 0, 0);
#else
  unsigned lds_off = (unsigned)(size_t)(l_v4i*)l;
  unsigned long long ga = (unsigned long long)(size_t)g;
  asm volatile("global_load_async_to_lds_b128 %0, %1, off"
               :: "v"(lds_off), "v"(ga)
               : "memory");
#endif
}

__device__ __forceinline__ void dtd_async_wait0() {
#ifdef DTD_USE_ASYNC_BUILTIN
  __builtin_amdgcn_s_wait_asynccnt(0);
#else
  asm volatile("s_wait_asynccnt 0" ::: "memory");
#endif
}

// ---------------------------------------------------------------------------
// Kernel 1: NCHW [4,128,30720] -> NHWC [4,30720,128] via padded LDS tile.
// Block = 256 threads, tile = 64 pixels x 128 channels.
// Load phase: async b128, coalesced along pixels. Store: coalesced along C.
// ---------------------------------------------------------------------------
__global__ void __launch_bounds__(256)
dtd_transpose_kernel(const float* __restrict__ src, float* __restrict__ dst) {
  // row stride 68 floats: keeps 16B alignment for b128 async, breaks bank collisions
  __shared__ float tile[kC][68];
  const int tid   = threadIdx.x;
  const int n     = blockIdx.y;          // image 0..3
  const int pbase = blockIdx.x * 64;     // pixel tile base (480 tiles)

  const float* sn = src + (size_t)n * kC * kP;
#pragma unroll
  for (int k = 0; k < 8; ++k) {
    const int i  = k * 256 + tid;        // 0..2047 float4 slots (128c x 16 groups)
    const int c  = i >> 4;
    const int p4 = (i & 15) * 4;
    dtd_async_copy_b128(sn + (size_t)c * kP + pbase + p4, &tile[c][p4]);
  }
  dtd_async_wait0();
  __syncthreads();

  float* dn = dst + ((size_t)n * kP + pbase) * kC;
#pragma unroll
  for (int k = 0; k < 32; ++k) {
    const int j = k * 256 + tid;         // 0..8191 (64p x 128c)
    const int p = j >> 7;
    const int c = j & 127;
    dn[(size_t)p * kC + c] = tile[c][p]; // 512B/row coalesced, conflict-light LDS read
  }
}

// ---------------------------------------------------------------------------
// Kernel 2: one wave32 per (b,q). Lane carries channels [4*lane .. 4*lane+3].
// All per-query scalar math is wave-uniform -> gather branches are uniform.
// NHWC=true reads the transposed image from d_ws; false reads original NCHW.
// ---------------------------------------------------------------------------
template <bool NHWC>
__global__ void __launch_bounds__(256)
dtd_sample_kernel(const float* __restrict__ mem,
                  const float* __restrict__ rp,
                  const float* __restrict__ Tmat,
                  const float* __restrict__ intr,
                  const float* __restrict__ ooi,
                  float* __restrict__ out) {
  const int wave = (int)((blockIdx.x * 256u + threadIdx.x) >> 5);
  const int lane = (int)(threadIdx.x & 31u);
  const int b = wave >> 17;          // kQ == 2^17
  const int q = wave & (kQ - 1);

  const float* rpp = rp + (size_t)(b * kQ + q) * 3;
  const float px = fmaf(rpp[0], 16.f, -8.f);
  const float py = fmaf(rpp[1], 16.f, -8.f);
  const float pz = fmaf(rpp[2], 32.f,  2.f);

  float o0, o1, o2, o3;
  if (NHWC) {
    const v4f o = *(const v4f*)(ooi + lane * 4);
    o0 = o.x; o1 = o.y; o2 = o.z; o3 = o.w;
  } else {
    o0 = ooi[lane]; o1 = ooi[lane + 32]; o2 = ooi[lane + 64]; o3 = ooi[lane + 96];
  }

  float acc0 = 0.f, acc1 = 0.f, acc2 = 0.f, acc3 = 0.f;

#pragma unroll
  for (int v = 0; v < kNV; ++v) {
    const int n = b * kNV + v;
    const float* t16 = Tmat + n * 16;
    const float r00 = t16[0], r01 = t16[1], r02 = t16[2],  tx = t16[3];
    const float r10 = t16[4], r11 = t16[5], r12 = t16[6],  ty = t16[7];
    const float r20 = t16[8], r21 = t16[9], r22 = t16[10], tz = t16[11];
    const float* it = intr + n * 4;
    const float fx = it[0], fy = it[1], cx = it[2], cy = it[3];

    const float pcx = fmaf(r00, px, fmaf(r01, py, fmaf(r02, pz, tx)));
    const float pcy = fmaf(r10, px, fmaf(r11, py, fmaf(r12, pz, ty)));
    const float pcz = fmaf(r20, px, fmaf(r21, py, fmaf(r22, pz, tz)));
    const float invz = 1.f / fmaxf(pcz, 1e-6f);
    const float u  = fmaf(fx * pcx, invz, cx);   // == sample x (align_corners)
    const float vv = fmaf(fy * pcy, invz, cy);   // == sample y
    const bool valid = (pcz > 0.001f) && (u >= 0.f) && (u <= (float)(kW - 1)) &&
                       (vv >= 0.f) && (vv <= (float)(kH - 1));

    if (lane == 0) {
      float* ci = out + kOffCenter + ((size_t)n * kQ + q) * 2;
      ci[0] = u;
      ci[1] = vv;
      out[kOffValid + (size_t)n * kQ + q] = valid ? 1.f : 0.f;
    }

    if (valid) {  // wave-uniform branch
      const float x0f = floorf(u), y0f = floorf(vv);
      const float wx1 = u - x0f, wx0 = 1.f - wx1;
      const float wy1 = vv - y0f, wy0 = 1.f - wy1;
      const int x0 = (int)x0f, y0 = (int)y0f, x1 = x0 + 1, y1 = y0 + 1;
      if (NHWC) {
        const float* bn = mem + (size_t)n * kP * kC + (size_t)lane * 4;
        auto corner = [&](int xi, int yi, float w) {
          const v4f cv = *(const v4f*)(bn + (size_t)(yi * kW + xi) * kC);
          acc0 = fmaf(w, cv.x, acc0); acc1 = fmaf(w, cv.y, acc1);
          acc2 = fmaf(w, cv.z, acc2); acc3 = fmaf(w, cv.w, acc3);
        };
        corner(x0, y0, wx0 * wy0);                       // always in-bounds when valid
        if (x1 < kW)             corner(x1, y0, wx1 * wy0);
        if (y1 < kH)             corner(x0, y1, wx0 * wy1);
        if (x1 < kW && y1 < kH)  corner(x1, y1, wx1 * wy1);
      } else {
        const float* bn = mem + (size_t)n * kC * kP;
        auto corner = [&](int xi, int yi, float w) {
          const size_t pix = (size_t)(yi * kW + xi);
          acc0 = fmaf(w, bn[(size_t)(lane      ) * kP + pix], acc0);
          acc1 = fmaf(w, bn[(size_t)(lane + 32 ) * kP + pix], acc1);
          acc2 = fmaf(w, bn[(size_t)(lane + 64 ) * kP + pix], acc2);
          acc3 = fmaf(w, bn[(size_t)(lane + 96 ) * kP + pix], acc3);
        };
        corner(x0, y0, wx0 * wy0);
        if (x1 < kW)             corner(x1, y0, wx1 * wy0);
        if (y1 < kH)             corner(x0, y1, wx0 * wy1);
        if (x1 < kW && y1 < kH)  corner(x1, y1, wx1 * wy1);
      }
    } else {
      acc0 += o0; acc1 += o1; acc2 += o2; acc3 += o3;
    }
  }

  const float s = 1.f / (float)kNV;
  if (NHWC) {
    v4f r;
    r.x = acc0 * s; r.y = acc1 * s; r.z = acc2 * s; r.w = acc3 * s;
    // non-temporal: don't let the 134MB feature stream evict the L2-resident image
    __builtin_nontemporal_store(r, (v4f*)(out + (size_t)(b * kQ + q) * kC + lane * 4));
  } else {
    float* op = out + (size_t)(b * kQ + q) * kC + lane;
    op[0]  = acc0 * s;
    op[32] = acc1 * s;
    op[64] = acc2 * s;
    op[96] = acc3 * s;
  }
}

// ---------------------------------------------------------------------------
// Host entry
// ---------------------------------------------------------------------------
extern "C" void kernel_launch(void* const* d_in, const int* in_sizes, int n_in,
                              void* d_out, int out_size, void* d_ws, size_t ws_size,
                              hipStream_t stream) {
  (void)in_sizes; (void)n_in; (void)out_size;
  const float* mem  = (const float*)d_in[0];  // [4,128,128,240]
  const float* rp   = (const float*)d_in[1];  // [2,131072,3]
  const float* Tm   = (const float*)d_in[2];  // [2,2,4,4]
  const float* intr = (const float*)d_in[3];  // [2,2,4]
  const float* ooi  = (const float*)d_in[4];  // [1,128,1,1]
  float* out = (float*)d_out;

  const size_t need = (size_t)kBS * kNV * kP * kC * sizeof(float);  // 62.9 MB NHWC copy
  const int sample_blocks = (kBS * kQ) / 8;                         // 8 waves per block

  if (d_ws != nullptr && ws_size >= need) {
    float* nhwc = (float*)d_ws;
    dim3 tg(kP / 64, kBS * kNV);  // (480, 4)
    dtd_transpose_kernel<<<tg, 256, 0, stream>>>(mem, nhwc);
    dtd_sample_kernel<true><<<sample_blocks, 256, 0, stream>>>(nhwc, rp, Tm, intr, ooi, out);
  } else {
    dtd_sample_kernel<false><<<sample_blocks, 256, 0, stream>>>(mem, rp, Tm, intr, ooi, out);
  }
}